// EvolveGCN_88768384074112
// MI455X (gfx1250) — compile-verified
//
#include <hip/hip_runtime.h>

#define C_DIM 768
#define GC    2304   // 3*C
#define SORTN 4096

typedef __attribute__((ext_vector_type(2))) float v2f;
typedef __attribute__((ext_vector_type(8))) float v8f;

__device__ __forceinline__ unsigned fkey(float f) {
  unsigned u = __float_as_uint(f);
  return u ^ ((u & 0x80000000u) ? 0xFFFFFFFFu : 0x80000000u); // order-preserving
}
__device__ __forceinline__ float sigm(float x) { return 1.0f / (1.0f + expf(-x)); }

// ---- normalize p: pn = p / ||p|| -----------------------------------------
__global__ void k_normp(const float* __restrict__ p, float* __restrict__ pn) {
  __shared__ float wsum[8];
  __shared__ float inv;
  int tid = threadIdx.x;
  float s = 0.f;
  for (int i = tid; i < C_DIM; i += 256) { float v = p[i]; s += v * v; }
  for (int o = 16; o; o >>= 1) s += __shfl_xor(s, o, 32);
  if ((tid & 31) == 0) wsum[tid >> 5] = s;
  __syncthreads();
  if (tid == 0) { float t = 0.f; for (int i = 0; i < 8; ++i) t += wsum[i]; inv = rsqrtf(t); }
  __syncthreads();
  for (int i = tid; i < C_DIM; i += 256) pn[i] = p[i] * inv;
}

// ---- score = tanh(X . pn), one wave per row ------------------------------
__global__ void k_score(const float* __restrict__ X, const float* __restrict__ pn,
                        float* __restrict__ score, int N) {
  int w = (int)((blockIdx.x * blockDim.x + threadIdx.x) >> 5);
  int lane = threadIdx.x & 31;
  if (w >= N) return;
  const float* x = X + (size_t)w * C_DIM;
  float s = 0.f;
  for (int i = lane; i < C_DIM; i += 32) s += x[i] * pn[i];
  for (int o = 16; o; o >>= 1) s += __shfl_xor(s, o, 32);
  if (lane == 0) score[w] = tanhf(s);
}

// ---- exact K-th largest key via 32-pass radix select (1 block) -----------
__global__ void k_select(const float* __restrict__ score, int N, int K,
                         unsigned* __restrict__ meta) {
  __shared__ int tot;
  int tid = threadIdx.x;
  unsigned prefix = 0; int rem = K;
  for (int b = 31; b >= 0; --b) {
    if (tid == 0) tot = 0;
    __syncthreads();
    unsigned want = (prefix << 1) | 1u;
    int c = 0;
    for (int i = tid; i < N; i += 1024) c += ((fkey(score[i]) >> b) == want);
    for (int o = 16; o; o >>= 1) c += __shfl_xor(c, o, 32);
    if ((tid & 31) == 0) atomicAdd(&tot, c);
    __syncthreads();
    int total = tot;
    if (total >= rem) prefix = want; else { rem -= total; prefix <<= 1; }
    __syncthreads();
  }
  if (tid == 0) { meta[0] = prefix; meta[1] = 0u; }
}

// ---- collect all candidates with key >= threshold ------------------------
__global__ void k_collect(const float* __restrict__ score, int N,
                          unsigned* __restrict__ meta, unsigned* __restrict__ cand) {
  int i = blockIdx.x * blockDim.x + threadIdx.x;
  if (i >= N) return;
  if (fkey(score[i]) >= meta[0]) {
    unsigned pos = atomicAdd(&meta[1], 1u);
    if (pos < SORTN) cand[pos] = (unsigned)i;
  }
}

// ---- bitonic sort candidates by (value desc, index asc); emit top 768 ----
__global__ void k_sort(const float* __restrict__ score, const unsigned* __restrict__ meta,
                       const unsigned* __restrict__ cand,
                       unsigned* __restrict__ perm, float* __restrict__ vals) {
  __shared__ unsigned long long keys[SORTN];
  int tid = threadIdx.x;
  int cnt = (int)meta[1]; if (cnt > SORTN) cnt = SORTN;
  for (int i = tid; i < SORTN; i += 1024) {
    if (i < cnt) {
      unsigned idx = cand[i];
      unsigned u = fkey(score[idx]);
      keys[i] = (((unsigned long long)(~u)) << 32) | (unsigned long long)idx;
    } else keys[i] = ~0ull;
  }
  __syncthreads();
  for (int sz = 2; sz <= SORTN; sz <<= 1) {
    for (int st = sz >> 1; st > 0; st >>= 1) {
      for (int i = tid; i < SORTN; i += 1024) {
        int j = i ^ st;
        if (j > i) {
          bool up = ((i & sz) == 0);
          unsigned long long a = keys[i], b = keys[j];
          if ((a > b) == up) { keys[i] = b; keys[j] = a; }
        }
      }
      __syncthreads();
    }
  }
  for (int i = tid; i < C_DIM; i += 1024) {
    unsigned idx = (unsigned)(keys[i] & 0xFFFFFFFFull);
    perm[i] = idx; vals[i] = score[idx];
  }
}

// ---- X_tilde[j,:] = X[perm[j],:] * vals[j] -------------------------------
__global__ void k_xtilde(const float* __restrict__ X, const unsigned* __restrict__ perm,
                         const float* __restrict__ vals, float* __restrict__ Xt) {
  int i = blockIdx.x * blockDim.x + threadIdx.x;
  if (i >= C_DIM * C_DIM) return;
  int r = i / C_DIM, c = i % C_DIM;
  Xt[i] = X[(size_t)perm[r] * C_DIM + c] * vals[r];
}

// ---- GRU elementwise; writes W TRANSPOSED: WtT[c*C + m] = W[m][c] --------
__global__ void k_gru(const float* __restrict__ gi, const float* __restrict__ gh,
                      const float* __restrict__ iw, float* __restrict__ WtT) {
  int i = blockIdx.x * blockDim.x + threadIdx.x;
  if (i >= C_DIM * C_DIM) return;
  int m = i / C_DIM, c = i % C_DIM;
  size_t base = (size_t)m * GC + c;
  float r = sigm(gi[base] + gh[base]);
  float z = sigm(gi[base + C_DIM] + gh[base + C_DIM]);
  float n = tanhf(gi[base + 2 * C_DIM] + r * gh[base + 2 * C_DIM]);
  WtT[(size_t)c * C_DIM + m] = (1.0f - z) * n + z * iw[i];
}

// ---- degree / dinv -------------------------------------------------------
__global__ void k_deginit(float* __restrict__ deg, int N) {
  int i = blockIdx.x * blockDim.x + threadIdx.x;
  if (i < N) deg[i] = 1.0f;  // self-loop
}
__global__ void k_degacc(const int* __restrict__ col, int Et, float* __restrict__ deg) {
  int e = blockIdx.x * blockDim.x + threadIdx.x;
  if (e < Et) unsafeAtomicAdd(&deg[col[e]], 1.0f);  // exact integer-valued adds
}
__global__ void k_dinv(float* __restrict__ deg, int N) {
  int i = blockIdx.x * blockDim.x + threadIdx.x;
  if (i < N) { float d = deg[i]; deg[i] = (d > 0.f) ? rsqrtf(d) : 0.f; }
}

// ---- Y init with self-loop term: Y[i,:] = dinv[i]^2 * X[i,:] -------------
__global__ void k_yinit(const float* __restrict__ X, const float* __restrict__ dinv,
                        float* __restrict__ Y, long long total) {
  long long i = (long long)blockIdx.x * blockDim.x + threadIdx.x;
  if (i >= total) return;
  int r = (int)(i / C_DIM);
  float dv = dinv[r];
  Y[i] = X[i] * dv * dv;
}

// ---- edge scatter: Y[col,:] += dinv[row]*dinv[col] * X[row,:] (1 wave/edge)
__global__ void k_scatter(const float* __restrict__ X, const float* __restrict__ dinv,
                          const int* __restrict__ row, const int* __restrict__ col,
                          int Et, float* __restrict__ Y) {
  int w = (int)((blockIdx.x * blockDim.x + threadIdx.x) >> 5);
  int lane = threadIdx.x & 31;
  if (w >= Et) return;
  int r = row[w], c = col[w];
  float nrm = dinv[r] * dinv[c];
  const float* xs = X + (size_t)r * C_DIM;
  float* yd = Y + (size_t)c * C_DIM;
  for (int i = lane; i < C_DIM; i += 32) unsafeAtomicAdd(&yd[i], nrm * xs[i]);
}

// ---- fp32 WMMA GEMM: C(MxN) = A(MxK) * Bt(NxK)^T (+bias) -----------------
// B is ALWAYS given row-major as (N x K) ("B-transposed" storage).
// Block: 256 threads (8 waves as 4m x 2n), tile 64(M) x 128(N), K chunk 32.
// LDS is k-minor for BOTH operands so every fragment load merges into b64.
__global__ void k_gemm(const float* __restrict__ A, const float* __restrict__ Bt,
                       const float* __restrict__ bias, float* __restrict__ Cm,
                       int M, int Nn, int K) {
  __shared__ __align__(16) float As[64][36];    // [m][k], 32 cols + pad
  __shared__ __align__(16) float Bs[128][36];   // [n][k], 32 cols + pad
  int tid = threadIdx.x;
  int wid = tid >> 5, lane = tid & 31;
  int wm = wid >> 1, wn = wid & 1;              // 4x2 wave grid
  int m0 = blockIdx.x * 64, n0 = blockIdx.y * 128;
  int ln = lane & 15;
  int kb = (lane >> 4) << 1;                    // 0 or 2: K sub-pair per lane half
  v8f acc[4] = {};
  for (int k0 = 0; k0 < K; k0 += 32) {
    // stage A tile 64x32 via float4 (row-major copy)
    for (int i = tid; i < 64 * 8; i += 256) {
      int r = i >> 3, c4 = (i & 7) << 2;
      int gr = m0 + r;
      float4 v = make_float4(0.f, 0.f, 0.f, 0.f);
      if (gr < M) v = *(const float4*)&A[(size_t)gr * K + k0 + c4];
      *(float4*)&As[r][c4] = v;
    }
    // stage B tile 128x32 via float4 (row-major copy of (N x K) storage)
    for (int i = tid; i < 128 * 8; i += 256) {
      int n = i >> 3, c4 = (i & 7) << 2;
      *(float4*)&Bs[n][c4] = *(const float4*)&Bt[(size_t)(n0 + n) * K + k0 + c4];
    }
    __syncthreads();
#pragma unroll
    for (int kk = 0; kk < 32; kk += 4) {
      v2f a;
      a.x = As[wm * 16 + ln][kk + kb];
      a.y = As[wm * 16 + ln][kk + kb + 1];
#pragma unroll
      for (int nt = 0; nt < 4; ++nt) {
        v2f b;
        b.x = Bs[wn * 64 + nt * 16 + ln][kk + kb];
        b.y = Bs[wn * 64 + nt * 16 + ln][kk + kb + 1];
        acc[nt] = __builtin_amdgcn_wmma_f32_16x16x4_f32(false, a, false, b,
                                                        (short)0, acc[nt], false, false);
      }
    }
    __syncthreads();
  }
  // C/D 16x16 f32: VGPR g -> M = g (+8 for lanes 16-31), N = lane&15
  int moff = (lane >> 4) * 8;
#pragma unroll
  for (int nt = 0; nt < 4; ++nt) {
    int nc = n0 + wn * 64 + nt * 16 + ln;
    float bv = bias ? bias[nc] : 0.0f;
#pragma unroll
    for (int g = 0; g < 8; ++g) {
      int gr = m0 + wm * 16 + g + moff;
      if (gr < M) Cm[(size_t)gr * Nn + nc] = acc[nt][g] + bv;
    }
  }
}

extern "C" void kernel_launch(void* const* d_in, const int* in_sizes, int n_in,
                              void* d_out, int out_size, void* d_ws, size_t ws_size,
                              hipStream_t stream) {
  const float* X0  = (const float*)d_in[0];
  const int*   ei  = (const int*)d_in[1];
  const float* iw  = (const float*)d_in[2];
  const float* p   = (const float*)d_in[3];
  const float* Wih = (const float*)d_in[4];
  const float* Whh = (const float*)d_in[5];
  const float* bih = (const float*)d_in[6];
  const float* bhh = (const float*)d_in[7];
  int N  = in_sizes[0] / C_DIM;
  int E  = in_sizes[1] / 2;
  const int T = 4;
  int Et = E / T;

  char* w = (char*)d_ws;
  auto carve = [&](size_t bytes) -> char* {
    char* r = w; w += (bytes + 255) & ~(size_t)255; return r;
  };
  float*    pn   = (float*)carve((size_t)C_DIM * 4);
  float*    gh   = (float*)carve((size_t)C_DIM * GC * 4);
  float*    gi   = (float*)carve((size_t)C_DIM * GC * 4);
  float*    scr  = (float*)carve((size_t)N * 4);
  float*    deg  = (float*)carve((size_t)N * 4);
  unsigned* meta = (unsigned*)carve(64);
  unsigned* cand = (unsigned*)carve((size_t)SORTN * 4);
  unsigned* perm = (unsigned*)carve((size_t)C_DIM * 4);
  float*    vals = (float*)carve((size_t)C_DIM * 4);
  float*    Xt   = (float*)carve((size_t)C_DIM * C_DIM * 4);
  float*    WtT  = (float*)carve((size_t)C_DIM * C_DIM * 4);  // W transposed (N x K)
  float*    Y    = (float*)carve((size_t)N * C_DIM * 4);
  float*    Xb   = (float*)carve((size_t)N * C_DIM * 4);

  k_normp<<<1, 256, 0, stream>>>(p, pn);
  // gh depends only on initial_weight: hoist out of the time loop
  {
    dim3 g(C_DIM / 64, GC / 128);
    k_gemm<<<g, 256, 0, stream>>>(iw, Whh, bhh, gh, C_DIM, GC, C_DIM);
  }

  const float* cur = X0;
  for (int t = 0; t < T; ++t) {
    float* dst = (t == T - 1) ? (float*)d_out : Xb;
    const int* rowp = ei + (size_t)t * Et;
    const int* colp = ei + (size_t)E + (size_t)t * Et;

    k_score<<<(N + 7) / 8, 256, 0, stream>>>(cur, pn, scr, N);
    k_select<<<1, 1024, 0, stream>>>(scr, N, C_DIM, meta);
    k_collect<<<(N + 255) / 256, 256, 0, stream>>>(scr, N, meta, cand);
    k_sort<<<1, 1024, 0, stream>>>(scr, meta, cand, perm, vals);
    k_xtilde<<<(C_DIM * C_DIM + 255) / 256, 256, 0, stream>>>(cur, perm, vals, Xt);
    {
      dim3 g(C_DIM / 64, GC / 128);
      k_gemm<<<g, 256, 0, stream>>>(Xt, Wih, bih, gi, C_DIM, GC, C_DIM);
    }
    // k_gru writes W transposed so the big GEMM's B is (N x K) row-major
    k_gru<<<(C_DIM * C_DIM + 255) / 256, 256, 0, stream>>>(gi, gh, iw, WtT);

    k_deginit<<<(N + 255) / 256, 256, 0, stream>>>(deg, N);
    k_degacc<<<(Et + 255) / 256, 256, 0, stream>>>(colp, Et, deg);
    k_dinv<<<(N + 255) / 256, 256, 0, stream>>>(deg, N);

    long long tot = (long long)N * C_DIM;
    k_yinit<<<(unsigned)((tot + 255) / 256), 256, 0, stream>>>(cur, deg, Y, tot);
    k_scatter<<<(Et + 7) / 8, 256, 0, stream>>>(cur, deg, rowp, colp, Et, Y);
    {
      // propagate-then-multiply: out = (norm-propagated X) @ W  (linearity)
      dim3 g((N + 63) / 64, C_DIM / 128);
      k_gemm<<<g, 256, 0, stream>>>(Y, WtT, nullptr, dst, N, C_DIM, C_DIM);
    }
    cur = dst;
  }
  (void)n_in; (void)out_size; (void)ws_size;
}